// LCALayer_27771258536509
// MI455X (gfx1250) — compile-verified
//
#include <hip/hip_runtime.h>
#include <math.h>

// LCA hyperparameters (from the reference)
#define THRESHOLD 1.0f
#define LEAK      0.1f
#define COMPETITION 0.1f
#define DT        0.01f
#define SQRT_STEP 0.03162277660168379f   // sqrt(0.001), hardcoded in the original module
#define NDIM      8

typedef float v2f __attribute__((ext_vector_type(2)));
typedef float v8f __attribute__((ext_vector_type(8)));

// Main kernel: 256 threads = 8 wave32; each wave handles 16 sims.
// Lane n (0..15) owns dims {0,1,4,5} of sim base+n; lane n+16 owns dims {2,3,6,7}.
// This per-lane b64 chunking IS the f32 WMMA B-matrix layout, so the activity
// registers feed V_WMMA_F32_16X16X4_F32 directly. With A == const(-0.7), the
// two chained WMMAs compute recur[n] = -0.7 * rowsum(activities[n]) and the
// C/D layout broadcasts recur to both lane halves in D's VGPR0 (c[0]).
__global__ __launch_bounds__(256) void lca_wmma_kernel(
    const float* __restrict__ inp,
    const float* __restrict__ pre_in,
    const float* __restrict__ acts,
    const float* __restrict__ noise,
    float* __restrict__ out_pre,
    float* __restrict__ out_act,
    float* __restrict__ out_active)
{
    const int  lane = threadIdx.x & 31;
    const int  wave = threadIdx.x >> 5;
    const int  half = lane >> 4;                       // 0 or 1 (which dim group)
    const int  n    = lane & 15;                       // sim index within wave tile
    const long sim  = (long)blockIdx.x * 128 + (long)wave * 16 + n;
    const long rb   = sim * NDIM + half * 2;           // element offset of first b64 chunk

    // ---- activities: double duty as WMMA B operands and the |.|<1 mask ----
    v2f b0 = __builtin_nontemporal_load((const v2f*)(acts + rb));       // K-dims {0,1}|{2,3}
    v2f b1 = __builtin_nontemporal_load((const v2f*)(acts + rb + 4));   // K-dims {4,5}|{6,7}

    // A-matrix (16x4, f32): every element = -(NDIM-1)*COMPETITION = -0.7
    v2f a;
    a.x = -(float)(NDIM - 1) * COMPETITION;
    a.y = a.x;

    v8f c = {};
    c = __builtin_amdgcn_wmma_f32_16x16x4_f32(false, a, false, b0, (short)0, c, false, false);
    c = __builtin_amdgcn_wmma_f32_16x16x4_f32(false, a, false, b1, (short)0, c, false, false);
    // D is rank-1 (all rows equal); VGPR0 holds recur[lane&15] in BOTH lane halves.
    const float recur = c[0];

    // active: all 8 dims below threshold -> AND my 4 dims with partner lane's 4
    int mine = (fabsf(b0.x) < THRESHOLD && fabsf(b0.y) < THRESHOLD &&
                fabsf(b1.x) < THRESHOLD && fabsf(b1.y) < THRESHOLD) ? 1 : 0;
    int both = mine & __shfl_xor(mine, 16, 32);
    const float active = both ? 1.0f : 0.0f;

    // ---- streaming elementwise update (NT hints: single-use data, skip L2 retention) ----
    v2f i0 = __builtin_nontemporal_load((const v2f*)(inp    + rb));
    v2f i1 = __builtin_nontemporal_load((const v2f*)(inp    + rb + 4));
    v2f p0 = __builtin_nontemporal_load((const v2f*)(pre_in + rb));
    v2f p1 = __builtin_nontemporal_load((const v2f*)(pre_in + rb + 4));
    v2f z0 = __builtin_nontemporal_load((const v2f*)(noise  + rb));
    v2f z1 = __builtin_nontemporal_load((const v2f*)(noise  + rb + 4));

    const float s  = active * DT;          // (.. )*active*time_step
    const float ns = active * SQRT_STEP;   // noise * (active*NOISE) * sqrt(0.001), NOISE==1

    v2f q0, q1, r0, r1;
    q0.x = fmaf(i0.x - LEAK * p0.x + recur, s, fmaf(z0.x, ns, p0.x));
    q0.y = fmaf(i0.y - LEAK * p0.y + recur, s, fmaf(z0.y, ns, p0.y));
    q1.x = fmaf(i1.x - LEAK * p1.x + recur, s, fmaf(z1.x, ns, p1.x));
    q1.y = fmaf(i1.y - LEAK * p1.y + recur, s, fmaf(z1.y, ns, p1.y));
    r0.x = fmaxf(q0.x, 0.0f);  r0.y = fmaxf(q0.y, 0.0f);
    r1.x = fmaxf(q1.x, 0.0f);  r1.y = fmaxf(q1.y, 0.0f);

    __builtin_nontemporal_store(q0, (v2f*)(out_pre + rb));
    __builtin_nontemporal_store(q1, (v2f*)(out_pre + rb + 4));
    __builtin_nontemporal_store(r0, (v2f*)(out_act + rb));
    __builtin_nontemporal_store(r1, (v2f*)(out_act + rb + 4));
    if (half == 0)                                  // 16 lanes -> one coalesced 64B burst
        __builtin_nontemporal_store(active, out_active + sim);
}

// Scalar tail for sim counts not divisible by 128 (not hit for N=2,000,000).
__global__ void lca_tail_kernel(
    const float* __restrict__ inp,
    const float* __restrict__ pre_in,
    const float* __restrict__ acts,
    const float* __restrict__ noise,
    float* __restrict__ out_pre,
    float* __restrict__ out_act,
    float* __restrict__ out_active,
    long start, long n_sims)
{
    long sim = start + (long)blockIdx.x * blockDim.x + threadIdx.x;
    if (sim >= n_sims) return;
    const long rb = sim * NDIM;
    float av[NDIM];
    float sum = 0.0f;
    bool ok = true;
    for (int k = 0; k < NDIM; ++k) {
        av[k] = acts[rb + k];
        sum += av[k];
        ok = ok && (fabsf(av[k]) < THRESHOLD);
    }
    const float recur  = -(float)(NDIM - 1) * COMPETITION * sum;
    const float active = ok ? 1.0f : 0.0f;
    const float s  = active * DT;
    const float ns = active * SQRT_STEP;
    for (int k = 0; k < NDIM; ++k) {
        float p = pre_in[rb + k];
        float q = fmaf(inp[rb + k] - LEAK * p + recur, s, fmaf(noise[rb + k], ns, p));
        out_pre[rb + k] = q;
        out_act[rb + k] = fmaxf(q, 0.0f);
    }
    out_active[sim] = active;
}

extern "C" void kernel_launch(void* const* d_in, const int* in_sizes, int n_in,
                              void* d_out, int out_size, void* d_ws, size_t ws_size,
                              hipStream_t stream)
{
    const float* inp    = (const float*)d_in[0];
    const float* pre_in = (const float*)d_in[1];
    const float* acts   = (const float*)d_in[2];
    const float* noise  = (const float*)d_in[3];

    const long n_sims = (long)in_sizes[0] / NDIM;

    float* out_pre    = (float*)d_out;
    float* out_act    = out_pre + n_sims * NDIM;
    float* out_active = out_act + n_sims * NDIM;

    const long main_sims = (n_sims / 128) * 128;   // 128 sims per 256-thread block
    if (main_sims > 0) {
        lca_wmma_kernel<<<(int)(main_sims / 128), 256, 0, stream>>>(
            inp, pre_in, acts, noise, out_pre, out_act, out_active);
    }
    const long rem = n_sims - main_sims;
    if (rem > 0) {
        lca_tail_kernel<<<(int)((rem + 255) / 256), 256, 0, stream>>>(
            inp, pre_in, acts, noise, out_pre, out_act, out_active, main_sims, n_sims);
    }
}